// OrientedPool_15195594293506
// MI455X (gfx1250) — compile-verified
//
#include <hip/hip_runtime.h>
#include <hip/hip_bf16.h>
#include <stdint.h>

typedef __attribute__((ext_vector_type(2))) float v2f;
typedef __attribute__((ext_vector_type(8))) float v8f;

#define GC_B     64
#define GC_N     1024
#define GC_D     512
#define GC_TOTAL (GC_B * GC_N)
#define GC_K     512           // ceil(0.5 * 1024)
#define GC_NSEL  (GC_B * GC_K)

// ---------------------------------------------------------------- zero scratch
__global__ void k_zero(float* __restrict__ p, int n) {
    int i = blockIdx.x * blockDim.x + threadIdx.x;
    if (i < n) p[i] = 0.0f;
}

// ---------------------------------------------------------------- degree atomics
__global__ void k_deg(const int* __restrict__ esrc, const int* __restrict__ edst,
                      float* __restrict__ deg_out, float* __restrict__ deg_in, int E) {
    int e = blockIdx.x * blockDim.x + threadIdx.x;
    if (e < E) {
        atomicAdd(&deg_out[esrc[e]], 1.0f);
        atomicAdd(&deg_in[edst[e]], 1.0f);
    }
}

// ---------------------------------------------------------------- WMMA f32 matvec
// One wave (32 threads) computes h[r] = rsqrt(max(outdeg,1)) * dot(feature[r,:], W)
// for 16 rows using V_WMMA_F32_16X16X4_F32.  B[k][n] = W[k] for all n, so every
// column of D holds the same exact-f32 dot product.
// A layout (16x4 f32): lanes 0-15 hold K=0 (v0) / K=1 (v1); lanes 16-31 hold K=2/K=3.
__global__ __launch_bounds__(32) void k_matvec(const float* __restrict__ feat,
                                               const float* __restrict__ W,
                                               const float* __restrict__ deg_out,
                                               float* __restrict__ h) {
    const int lane = threadIdx.x;
    const int r0   = blockIdx.x * 16;
    const int m    = lane & 15;
    const int koff = (lane >> 4) * 2;          // 0 for lanes 0-15, 2 for lanes 16-31
    const float* __restrict__ arow = feat + (size_t)(r0 + m) * GC_D;

    v8f c = {};
    for (int k0 = 0; k0 < GC_D; k0 += 4) {
        v2f a, bm;
        a.x  = arow[k0 + koff];
        a.y  = arow[k0 + koff + 1];
        bm.x = W[k0 + koff];
        bm.y = W[k0 + koff + 1];
        c = __builtin_amdgcn_wmma_f32_16x16x4_f32(
                /*neg_a=*/false, a, /*neg_b=*/false, bm,
                /*c_mod=*/(short)0, c, /*reuse_a=*/false, /*reuse_b=*/false);
    }
    // D layout: lanes 0-15 carry rows M=0..7 in c[0..7]; lanes 16-31 carry M=8..15.
    __shared__ float hs[16];
    if (lane == 0)  { for (int i = 0; i < 8; ++i) hs[i]     = c[i]; }
    if (lane == 16) { for (int i = 0; i < 8; ++i) hs[8 + i] = c[i]; }
    __syncthreads();
    if (lane < 16) {
        int r = r0 + lane;
        float d = deg_out[r];
        d = d < 1.0f ? 1.0f : d;
        h[r] = hs[lane] * rsqrtf(d);
    }
}

// ---------------------------------------------------------------- edge aggregate
__global__ void k_agg(const int* __restrict__ esrc, const int* __restrict__ edst,
                      const float* __restrict__ h, float* __restrict__ agg, int E) {
    int e = blockIdx.x * blockDim.x + threadIdx.x;
    if (e < E) atomicAdd(&agg[edst[e]], h[esrc[e]]);
}

// ---------------------------------------------------------------- per-graph top-k
// Full bitonic sort of 1024 (score, idx) pairs, descending, ties -> smaller idx
// first (matches jax.lax.top_k ordering).  Keep the first 512.
__global__ __launch_bounds__(1024) void k_topk(const float* __restrict__ agg,
                                               const float* __restrict__ deg_in,
                                               const float* __restrict__ bias,
                                               float* __restrict__ selsc,
                                               int* __restrict__ perm_ws,
                                               int* __restrict__ perm_out) {
    __shared__ float key[GC_N];
    __shared__ int   idx[GC_N];
    const int g = blockIdx.x, t = threadIdx.x;
    const int n = g * GC_N + t;
    float din = deg_in[n];
    din = din < 1.0f ? 1.0f : din;
    key[t] = agg[n] * rsqrtf(din) + bias[0];
    idx[t] = t;
    __syncthreads();

    for (int k = 2; k <= GC_N; k <<= 1) {
        for (int j = k >> 1; j > 0; j >>= 1) {
            int ixj = t ^ j;
            if (ixj > t) {
                float ka = key[t], kb = key[ixj];
                int   ia = idx[t], ib = idx[ixj];
                bool t_first = (ka > kb) || (ka == kb && ia < ib);  // descending order
                bool desc = ((t & k) == 0);
                if (desc ? !t_first : t_first) {
                    key[t] = kb; key[ixj] = ka;
                    idx[t] = ib; idx[ixj] = ia;
                }
            }
            __syncthreads();
        }
    }
    if (t < GC_K) {
        int p = g * GC_N + idx[t];
        perm_ws[g * GC_K + t]  = p;
        perm_out[g * GC_K + t] = p;      // second output (flat int32 bits)
        selsc[g * GC_K + t]    = key[t];
    }
}

// ---------------------------------------------------------------- gather + scale
// pooled[i,:] = feature[perm[i],:] * tanh(score[perm[i]])
// Each wave owns one 2KB row at a time, double-buffered in LDS via CDNA5 async
// global->LDS B128 transfers (tracked by ASYNCcnt).
__device__ __forceinline__ void issue_row_async(const float* __restrict__ grow,
                                                unsigned lds_base, int lane) {
    for (int inst = 0; inst < 4; ++inst) {
        unsigned     loff = lds_base + (unsigned)(inst * 512 + lane * 16);
        const float* gad  = grow + inst * 128 + lane * 4;
        asm volatile("global_load_async_to_lds_b128 %0, %1, off"
                     :: "v"(loff), "v"(gad) : "memory");
    }
}

__global__ __launch_bounds__(256) void k_gather(const float* __restrict__ feat,
                                                const int* __restrict__ perm,
                                                const float* __restrict__ selsc,
                                                float* __restrict__ out, int nrows) {
    __shared__ float tile[8][2][GC_D];           // 8 waves x 2 buffers x 2KB
    const int lane = threadIdx.x & 31;
    const int wid  = threadIdx.x >> 5;
    const int gw   = blockIdx.x * 8 + wid;       // global wave id
    const int nw   = gridDim.x * 8;

    int cur = 0;
    int r = gw;
    if (r < nrows) {
        issue_row_async(feat + (size_t)perm[r] * GC_D,
                        (unsigned)(uintptr_t)&tile[wid][0][0], lane);
    }
    for (; r < nrows; r += nw) {
        int rn = r + nw;                          // wave-uniform
        if (rn < nrows) {
            issue_row_async(feat + (size_t)perm[rn] * GC_D,
                            (unsigned)(uintptr_t)&tile[wid][cur ^ 1][0], lane);
            asm volatile("s_wait_asynccnt 4" ::: "memory");   // current row landed
        } else {
            asm volatile("s_wait_asynccnt 0" ::: "memory");
        }
        float sc = tanhf(selsc[r]);
        const float4* __restrict__ src = (const float4*)&tile[wid][cur][0];
        float4* __restrict__ dst = (float4*)(out + (size_t)r * GC_D);
        for (int i = lane; i < GC_D / 4; i += 32) {
            float4 v = src[i];
            v.x *= sc; v.y *= sc; v.z *= sc; v.w *= sc;
            dst[i] = v;
        }
        cur ^= 1;
    }
}

// ---------------------------------------------------------------- launch
extern "C" void kernel_launch(void* const* d_in, const int* in_sizes, int n_in,
                              void* d_out, int out_size, void* d_ws, size_t ws_size,
                              hipStream_t stream) {
    const float* feature = (const float*)d_in[0];
    const int*   esrc    = (const int*)d_in[1];
    const int*   edst    = (const int*)d_in[2];
    const float* W       = (const float*)d_in[3];
    const float* bias    = (const float*)d_in[4];
    const int    E       = in_sizes[1];

    // workspace layout (floats): deg_out | deg_in | agg | h | selsc | perm(int)
    float* ws      = (float*)d_ws;
    float* deg_out = ws;
    float* deg_in  = ws + GC_TOTAL;
    float* aggbuf  = ws + 2 * GC_TOTAL;
    float* hbuf    = ws + 3 * GC_TOTAL;
    float* selsc   = ws + 4 * GC_TOTAL;
    int*   perm_ws = (int*)(ws + 4 * GC_TOTAL + GC_NSEL);

    float* pooled   = (float*)d_out;                           // [NSEL, D] f32
    int*   perm_out = (int*)((float*)d_out + (size_t)GC_NSEL * GC_D);

    // 1. zero deg_out, deg_in, agg (contiguous)
    k_zero<<<(3 * GC_TOTAL + 255) / 256, 256, 0, stream>>>(deg_out, 3 * GC_TOTAL);
    // 2. degrees
    k_deg<<<(E + 255) / 256, 256, 0, stream>>>(esrc, edst, deg_out, deg_in, E);
    // 3. WMMA matvec: 16 rows per wave
    k_matvec<<<GC_TOTAL / 16, 32, 0, stream>>>(feature, W, deg_out, hbuf);
    // 4. edge aggregation
    k_agg<<<(E + 255) / 256, 256, 0, stream>>>(esrc, edst, hbuf, aggbuf, E);
    // 5. per-graph top-512 (bitonic), emits perm + selected scores
    k_topk<<<GC_B, GC_N, 0, stream>>>(aggbuf, deg_in, bias, selsc, perm_ws, perm_out);
    // 6. gather rows via async global->LDS, scale by tanh(score)
    k_gather<<<512, 256, 0, stream>>>(feature, perm_ws, selsc, pooled, GC_NSEL);
}